// NonMaximaSuppression2d_35021163332234
// MI455X (gfx1250) — compile-verified
//
#include <hip/hip_runtime.h>

#define TILE_H 16
#define W      512
#define HGT    512
#define NPLANE (4 * 64)

typedef __attribute__((ext_vector_type(4))) int   v4i;
typedef __attribute__((ext_vector_type(4))) float v4f;
typedef __attribute__((address_space(1))) v4i* gptr_v4i;
typedef __attribute__((address_space(3))) v4i* lptr_v4i;

#if defined(__AMDGCN__) && __has_builtin(__builtin_amdgcn_global_load_async_to_lds_b128)
#define USE_ASYNC_LDS 1
#else
#define USE_ASYNC_LDS 0
#endif

__device__ __forceinline__ void wait_async_zero() {
#if __has_builtin(__builtin_amdgcn_s_wait_asynccnt)
    __builtin_amdgcn_s_wait_asynccnt(0);
#else
    asm volatile("s_wait_asynccnt 0" ::: "memory");
#endif
}

__global__ __launch_bounds__(256) void nms2d_kernel(const float* __restrict__ x,
                                                    float* __restrict__ out) {
    // (TILE_H + 2) halo rows of a full-width 512-float row: 36 KB LDS.
    __shared__ float tile[(TILE_H + 2) * W];

    const int plane = blockIdx.y;                 // 0..255  (b*c planes)
    const int y0    = blockIdx.x * TILE_H;        // first output row of tile
    const float* __restrict__ src = x   + (size_t)plane * (W * HGT);
    float* __restrict__ dst       = out + (size_t)plane * (W * HGT);

    // ---- Stage tile (rows y0-1 .. y0+TILE_H, clamped) into LDS, b128/lane ----
    for (int i = threadIdx.x; i < (TILE_H + 2) * (W / 4); i += 256) {
        const int ly = i >> 7;              // tile row 0..17
        const int xb = (i & 127) << 2;      // column (float), multiple of 4
        int gy = y0 - 1 + ly;
        gy = gy < 0 ? 0 : (gy > HGT - 1 ? HGT - 1 : gy);
        const float* gp = src + (size_t)gy * W + xb;
        float* lp = &tile[ly * W + xb];
#if USE_ASYNC_LDS
        __builtin_amdgcn_global_load_async_to_lds_b128(
            (gptr_v4i)(v4i*)(void*)const_cast<float*>(gp),
            (lptr_v4i)(v4i*)(void*)lp,
            0, 0);
#else
        *(v4f*)lp = *(const v4f*)gp;
#endif
    }
#if USE_ASYNC_LDS
    wait_async_zero();
#endif
    __syncthreads();

    // ---- Compute: each thread produces 8 float4 outputs (TILE_H*128 / 256) ----
    for (int i = threadIdx.x; i < TILE_H * (W / 4); i += 256) {
        const int r  = i >> 7;              // tile row 0..15
        const int xb = (i & 127) << 2;      // output column (float), mult of 4
        const int gy = y0 + r;
        const int ly = r + 1;               // center row inside LDS tile

        const v4f up  = *(const v4f*)&tile[(ly - 1) * W + xb];
        const v4f ctr = *(const v4f*)&tile[ ly      * W + xb];
        const v4f dn  = *(const v4f*)&tile[(ly + 1) * W + xb];

        const int xl = (xb == 0)       ? 0       : xb - 1;
        const int xr = (xb == W - 4)   ? W - 1   : xb + 4;
        const float ul = tile[(ly - 1) * W + xl], ur = tile[(ly - 1) * W + xr];
        const float cl = tile[ ly      * W + xl], cr = tile[ ly      * W + xr];
        const float dl = tile[(ly + 1) * W + xl], dr = tile[(ly + 1) * W + xr];

        // 6-wide windows (cols xb-1 .. xb+4) for the three rows.
        const float a[6] = {ul, up.x,  up.y,  up.z,  up.w,  ur};
        const float m[6] = {cl, ctr.x, ctr.y, ctr.z, ctr.w, cr};
        const float b[6] = {dl, dn.x,  dn.y,  dn.z,  dn.w,  dr};

        float o[4];
#pragma unroll
        for (int j = 0; j < 4; ++j) {
            const float c = m[j + 1];
            const bool g = (c > m[j]) && (c > m[j + 2]) &&
                           (c > a[j]) && (c > a[j + 1]) && (c > a[j + 2]) &&
                           (c > b[j]) && (c > b[j + 1]) && (c > b[j + 2]);
            o[j] = g ? c : 0.0f;
        }

        // Border mask: first/last row and first/last column are always zero.
        if (gy == 0 || gy == HGT - 1) { o[0] = o[1] = o[2] = o[3] = 0.0f; }
        if (xb == 0)      o[0] = 0.0f;
        if (xb == W - 4)  o[3] = 0.0f;

        // Streaming output: non-temporal store (TH=NT) so the 256 MB write
        // stream doesn't evict reusable input halo lines from L2.
        v4f ov; ov.x = o[0]; ov.y = o[1]; ov.z = o[2]; ov.w = o[3];
        __builtin_nontemporal_store(ov, (v4f*)(dst + (size_t)gy * W + xb));
    }
}

extern "C" void kernel_launch(void* const* d_in, const int* in_sizes, int n_in,
                              void* d_out, int out_size, void* d_ws, size_t ws_size,
                              hipStream_t stream) {
    const float* x = (const float*)d_in[0];
    float* out     = (float*)d_out;
    dim3 grid(HGT / TILE_H, NPLANE);   // 32 x 256 = 8192 blocks
    nms2d_kernel<<<grid, 256, 0, stream>>>(x, out);
}